// GCN_original_37194416783379
// MI455X (gfx1250) — compile-verified
//
#include <hip/hip_runtime.h>
#include <hip/hip_bf16.h>

typedef float v2f __attribute__((ext_vector_type(2)));
typedef float v8f __attribute__((ext_vector_type(8)));

#define NN 100000   // nodes
#define NE 1200000  // edges (without self loops)
#define NF 128      // input features
#define NH 64       // hidden
#define NG 256      // graphs

// ---------------------------------------------------------------- utilities
__global__ void k_zero4(float4* __restrict__ p, int n4) {
  int i = blockIdx.x * blockDim.x + threadIdx.x;
  if (i < n4) p[i] = make_float4(0.f, 0.f, 0.f, 0.f);
}

// deg[dst] counting over real edges (self loop folded in k_dinv)
__global__ void k_deg(const int* __restrict__ dst, float* __restrict__ deg) {
  int e = blockIdx.x * blockDim.x + threadIdx.x;
  if (e < NE) atomicAdd(&deg[dst[e]], 1.0f);
}

__global__ void k_dinv(float* __restrict__ deg) {
  int i = blockIdx.x * blockDim.x + threadIdx.x;
  if (i < NN) {
    float d = deg[i] + 1.0f;       // + self loop; always >= 1
    deg[i] = 1.0f / sqrtf(d);
  }
}

// ------------------------------------------------- fp32 WMMA GEMM  C = A @ W
// A: [M,K] row-major, W: [K,64] row-major, C: [M,64] row-major.
// One wave -> 16x64 tile via 4x V_WMMA_F32_16X16X4_F32 per K-step of 4.
//
// W is staged in LDS in fragment-ready form: pair-row kp (= k/2) stores
// {W[2kp][n], W[2kp+1][n]} at sW[kp*PSTRIDE + 2n].  One ds_load_b64 per B
// fragment, no register shuffles.  PSTRIDE=160 dwords (%64 == 32) puts the
// two half-waves (reading kp and kp+1) on disjoint LDS bank halves.
template<int K>
__global__ __launch_bounds__(256)
void k_gemm_wmma(const float* __restrict__ A, const float* __restrict__ W,
                 float* __restrict__ C, int M) {
  constexpr int PSTRIDE = 160;
  __shared__ float sW[(K / 2) * PSTRIDE];
  for (int pos = threadIdx.x; pos < (K / 2) * PSTRIDE; pos += 256) {
    int kp  = pos / PSTRIDE;
    int rem = pos - kp * PSTRIDE;
    if (rem < 128) {                       // pads stay unwritten/unread
      int n  = rem >> 1;
      int kk = 2 * kp + (rem & 1);
      sW[pos] = W[kk * 64 + n];
    }
  }
  __syncthreads();

  const int wave = threadIdx.x >> 5;
  const int lane = threadIdx.x & 31;
  const int half = lane >> 4;     // 0: lanes 0-15, 1: lanes 16-31
  const int lrow = lane & 15;
  const int rowBase = blockIdx.x * 128 + wave * 16;
  if (rowBase >= M) return;       // wave-uniform (M % 16 == 0): EXEC stays all-ones

  // 16x4 f32 A fragment: lanes 0-15 hold K={k0,k0+1}, lanes 16-31 hold K={k0+2,k0+3}
  const float* __restrict__ arow = A + (size_t)(rowBase + lrow) * K;

  v8f acc0 = {}, acc1 = {}, acc2 = {}, acc3 = {};
  for (int k0 = 0; k0 < K; k0 += 4) {
    float2 av = *(const float2*)(arow + k0 + 2 * half);   // global_load_b64
    v2f a; a.x = av.x; a.y = av.y;

    // fragment-ready B: one aligned ds_load_b64 per 16-col block
    const v2f* bp = (const v2f*)&sW[(k0 / 2 + half) * PSTRIDE + 2 * lrow];
    v2f b0 = bp[0];
    v2f b1 = bp[16];   // +16 cols
    v2f b2 = bp[32];   // +32 cols
    v2f b3 = bp[48];   // +48 cols

    acc0 = __builtin_amdgcn_wmma_f32_16x16x4_f32(false, a, false, b0, (short)0, acc0, false, false);
    acc1 = __builtin_amdgcn_wmma_f32_16x16x4_f32(false, a, false, b1, (short)0, acc1, false, false);
    acc2 = __builtin_amdgcn_wmma_f32_16x16x4_f32(false, a, false, b2, (short)0, acc2, false, false);
    acc3 = __builtin_amdgcn_wmma_f32_16x16x4_f32(false, a, false, b3, (short)0, acc3, false, false);
  }

  // C/D layout: VGPR r -> M = rowBase + r + 8*half, N = nb*16 + lrow
  float* __restrict__ crow = C + (size_t)rowBase * 64;
#pragma unroll
  for (int r = 0; r < 8; ++r) {
    const int mo = (r + 8 * half) * 64 + lrow;
    crow[mo +  0] = acc0[r];
    crow[mo + 16] = acc1[r];
    crow[mo + 32] = acc2[r];
    crow[mo + 48] = acc3[r];
  }
}

// --------------------------------------- edge gather/scatter (L2-resident)
// one thread per (edge, 4 features): out[dst,f..f+3] += nrm * h[src,f..f+3]
__global__ void k_edge_scatter(const int* __restrict__ ei, const float* __restrict__ h,
                               const float* __restrict__ dinv, float* __restrict__ out) {
  int idx = blockIdx.x * blockDim.x + threadIdx.x;
  if (idx >= NE * 16) return;
  int e = idx >> 4;
  int f = (idx & 15) << 2;
  int s = ei[e];         // edge_index[0,:]
  int d = ei[NE + e];    // edge_index[1,:]
  float nrm = dinv[s] * dinv[d];
  float4 v = *(const float4*)(h + (size_t)s * 64 + f);    // global_load_b128
  float* o = out + (size_t)d * 64 + f;
  atomicAdd(o + 0, nrm * v.x);
  atomicAdd(o + 1, nrm * v.y);
  atomicAdd(o + 2, nrm * v.z);
  atomicAdd(o + 3, nrm * v.w);
}

// layer-1 epilogue: fold self-loop + bias + ReLU; writes back into h (elementwise)
__global__ void k_epilogue_relu(const float* __restrict__ agg, float* __restrict__ h,
                                const float* __restrict__ dinv, const float* __restrict__ b) {
  int idx = blockIdx.x * blockDim.x + threadIdx.x;
  if (idx >= NN * 16) return;
  int i = idx >> 4;
  int f = (idx & 15) << 2;
  float di = dinv[i];
  float s = di * di;
  float4 ag = *(const float4*)(agg + (size_t)idx * 4);
  float4 hv = *(const float4*)(h + (size_t)idx * 4);
  float4 bv = *(const float4*)(b + f);
  float4 r;
  r.x = fmaxf(ag.x + s * hv.x + bv.x, 0.0f);
  r.y = fmaxf(ag.y + s * hv.y + bv.y, 0.0f);
  r.z = fmaxf(ag.z + s * hv.z + bv.z, 0.0f);
  r.w = fmaxf(ag.w + s * hv.w + bv.w, 0.0f);
  *(float4*)(h + (size_t)idx * 4) = r;
}

// layer-2 epilogue fused into pooling: sums[batch[i]] += agg2 + dinv^2*h2
__global__ void k_pool_sum(const float* __restrict__ agg, const float* __restrict__ h,
                           const float* __restrict__ dinv, const int* __restrict__ batch,
                           float* __restrict__ sums) {
  int idx = blockIdx.x * blockDim.x + threadIdx.x;
  if (idx >= NN * 16) return;
  int i = idx >> 4;
  int f = (idx & 15) << 2;
  float di = dinv[i];
  float s = di * di;
  float4 ag = *(const float4*)(agg + (size_t)idx * 4);
  float4 hv = *(const float4*)(h + (size_t)idx * 4);
  float* o = sums + (size_t)batch[i] * 64 + f;
  atomicAdd(o + 0, ag.x + s * hv.x);
  atomicAdd(o + 1, ag.y + s * hv.y);
  atomicAdd(o + 2, ag.z + s * hv.z);
  atomicAdd(o + 3, ag.w + s * hv.w);
}

__global__ void k_pool_cnt(const int* __restrict__ batch, float* __restrict__ cnt) {
  int i = blockIdx.x * blockDim.x + threadIdx.x;
  if (i < NN) atomicAdd(&cnt[batch[i]], 1.0f);
}

// readout: out[g] = (sums[g]/cnt[g] + b2) @ Wl + bl   (b2 hoisted past the mean)
__global__ void k_final(const float* __restrict__ sums, const float* __restrict__ cnt,
                        const float* __restrict__ b2, const float* __restrict__ Wl,
                        const float* __restrict__ bl, float* __restrict__ out) {
  int g = threadIdx.x;
  if (g >= NG) return;
  float c = cnt[g];
  float acc = bl[0];
  if (c > 0.0f) {                 // empty graph -> pooled == 0 -> out = bl
    float inv = 1.0f / c;
    for (int f = 0; f < 64; ++f)
      acc += (sums[g * 64 + f] * inv + b2[f]) * Wl[f];
  }
  out[g] = acc;
}

// ----------------------------------------------------------------- driver
extern "C" void kernel_launch(void* const* d_in, const int* in_sizes, int n_in,
                              void* d_out, int out_size, void* d_ws, size_t ws_size,
                              hipStream_t stream) {
  const float* x     = (const float*)d_in[0];
  const int*   ei    = (const int*)d_in[1];
  const int*   batch = (const int*)d_in[2];
  const float* W1    = (const float*)d_in[3];
  const float* b1    = (const float*)d_in[4];
  const float* W2    = (const float*)d_in[5];
  const float* b2    = (const float*)d_in[6];
  const float* Wl    = (const float*)d_in[7];
  const float* bl    = (const float*)d_in[8];
  float* out = (float*)d_out;

  // workspace carve-up (~52 MB): dinv | bufA[NN,64] | bufB[NN,64] | sums | cnt
  float* ws   = (float*)d_ws;
  float* dinv = ws;
  float* bufA = dinv + NN;                   // 400000 B offset -> 16B aligned
  float* bufB = bufA + (size_t)NN * 64;
  float* sums = bufB + (size_t)NN * 64;
  float* cnt  = sums + NG * 64;

  const int T = 256;
  const int gN   = (NN + T - 1) / T;
  const int gE   = (NE + T - 1) / T;
  const int gNF4 = (NN * 16 + T - 1) / T;    // node-feature kernels, float4
  const int gEF4 = (NE * 16 + T - 1) / T;    // edge-feature kernels, float4
  const int gZ   = (NN * 16 + T - 1) / T;    // zeroing 6.4M floats as float4
  const int gMM  = (NN + 127) / 128;

  // degrees -> dinv
  k_zero4<<<(NN / 4 + T - 1) / T, T, 0, stream>>>((float4*)dinv, NN / 4);
  k_deg <<<gE, T, 0, stream>>>(ei + NE, dinv);
  k_dinv<<<gN, T, 0, stream>>>(dinv);

  // layer 1: h1 = x@W1 ; agg ; relu(agg + dinv^2*h1 + b1) -> bufA
  k_gemm_wmma<NF><<<gMM, 256, 0, stream>>>(x, W1, bufA, NN);
  k_zero4<<<gZ, T, 0, stream>>>((float4*)bufB, NN * 16);
  k_edge_scatter<<<gEF4, T, 0, stream>>>(ei, bufA, dinv, bufB);
  k_epilogue_relu<<<gNF4, T, 0, stream>>>(bufB, bufA, dinv, b1);

  // layer 2: h2 = relu@W2 -> bufB ; agg -> bufA
  k_gemm_wmma<NH><<<gMM, 256, 0, stream>>>(bufA, W2, bufB, NN);
  k_zero4<<<gZ, T, 0, stream>>>((float4*)bufA, NN * 16);
  k_edge_scatter<<<gEF4, T, 0, stream>>>(ei, bufB, dinv, bufA);

  // pooling + readout
  k_zero4<<<((NG * 64 + NG) / 4 + T - 1) / T, T, 0, stream>>>((float4*)sums, (NG * 64 + NG) / 4);
  k_pool_sum<<<gNF4, T, 0, stream>>>(bufA, bufB, dinv, batch, sums);
  k_pool_cnt<<<gN, T, 0, stream>>>(batch, cnt);
  k_final<<<1, 256, 0, stream>>>(sums, cnt, b2, Wl, bl, out);
}